// _KLD_MVNx2_17987323036508
// MI455X (gfx1250) — compile-verified
//
#include <hip/hip_runtime.h>
#include <hip/hip_bf16.h>

typedef float v2f __attribute__((ext_vector_type(2)));
typedef float v4f __attribute__((ext_vector_type(4)));
typedef float v8f __attribute__((ext_vector_type(8)));

#define D_DIM 1024
#define ROWS_PER_BLOCK 16
#define THREADS_PER_BLOCK (ROWS_PER_BLOCK * 32)   // one wave32 per row

__global__ __launch_bounds__(THREADS_PER_BLOCK)
void _KLD_MVNx2_kernel(const float* __restrict__ mu_q,
                       const float* __restrict__ sigma_q,
                       const float* __restrict__ mu_p,
                       const float* __restrict__ sigma_p,
                       float* __restrict__ out)
{
    __shared__ float partials[ROWS_PER_BLOCK * 4];

    const int lane = threadIdx.x & 31;
    const int wave = threadIdx.x >> 5;
    const int row  = blockIdx.x * ROWS_PER_BLOCK + wave;

    // Lane L covers elements [L*4 + 128*i, +4) : wave covers 512 contiguous
    // bytes per load instruction, fully coalesced streaming.
    const size_t base = (size_t)row * D_DIM + (size_t)lane * 4;
    const v4f* pq = (const v4f*)(mu_q    + base);
    const v4f* ps = (const v4f*)(sigma_q + base);
    const v4f* pp = (const v4f*)(mu_p    + base);
    const v4f* pt = (const v4f*)(sigma_p + base);

    float acc = 0.0f;
#pragma unroll
    for (int i = 0; i < D_DIM / (32 * 4); ++i) {
        // Non-temporal: 256MB single-use stream, keep it out of L2 residency.
        v4f mq = __builtin_nontemporal_load(pq + i * 32);
        v4f sq = __builtin_nontemporal_load(ps + i * 32);
        v4f mp = __builtin_nontemporal_load(pp + i * 32);
        v4f sp = __builtin_nontemporal_load(pt + i * 32);
#pragma unroll
        for (int j = 0; j < 4; ++j) {
            float diff = mq[j] - mp[j];
            // exp(sp-sq) + diff^2*exp(-sq) + (sq - sp); the "-1" per element
            // is folded into the final scale: 0.5*(total - D).
            acc += __expf(sp[j] - sq[j]) + diff * diff * __expf(-sq[j])
                 + (sq[j] - sp[j]);
        }
    }

    // Intra-wave reduce 32 -> 4 partials (land in lanes 0..3).
    acc += __shfl_down(acc, 16, 32);
    acc += __shfl_down(acc,  8, 32);
    acc += __shfl_down(acc,  4, 32);
    if (lane < 4) partials[wave * 4 + lane] = acc;
    __syncthreads();

    // Final cross-wave reduction via WMMA: A(16x4) = per-row partials,
    // B(4x16) = ones  =>  D[m][n] = rowsum(A[m]) for every n.
    // A layout (32-bit 16x4): lanes 0-15 hold {K0,K1} of row M=lane,
    // lanes 16-31 hold {K2,K3} of row M=lane-16.
    if (wave == 0) {
        const int m  = lane & 15;
        const int kb = (lane >> 4) << 1;          // 0 or 2
        v2f A;
        A.x = partials[m * 4 + kb + 0];
        A.y = partials[m * 4 + kb + 1];
        v2f Bm; Bm.x = 1.0f; Bm.y = 1.0f;         // all-ones B, layout-agnostic
        v8f C = {};
        C = __builtin_amdgcn_wmma_f32_16x16x4_f32(
                /*neg_a=*/false, A, /*neg_b=*/false, Bm,
                /*c_mod=*/(short)0, C, /*reuse_a=*/false, /*reuse_b=*/false);

        // C/D layout: lane 0 VGPR v = D[v][0] (rows 0..7),
        //             lane 16 VGPR v = D[v+8][0] (rows 8..15).
        if (lane == 0 || lane == 16) {
            float* o = out + (size_t)blockIdx.x * ROWS_PER_BLOCK
                           + ((lane >> 4) << 3);
            v4f lo, hi;
#pragma unroll
            for (int r = 0; r < 4; ++r) {
                lo[r] = 0.5f * (C[r]     - (float)D_DIM);
                hi[r] = 0.5f * (C[r + 4] - (float)D_DIM);
            }
            *(v4f*)(o)     = lo;   // two b128 stores per writer lane
            *(v4f*)(o + 4) = hi;
        }
    }
}

extern "C" void kernel_launch(void* const* d_in, const int* in_sizes, int n_in,
                              void* d_out, int out_size, void* d_ws, size_t ws_size,
                              hipStream_t stream) {
    const float* mu_q    = (const float*)d_in[0];
    const float* sigma_q = (const float*)d_in[1];
    const float* mu_p    = (const float*)d_in[2];
    const float* sigma_p = (const float*)d_in[3];
    float* out = (float*)d_out;

    const int B = out_size;                    // 16384 rows
    const int blocks = B / ROWS_PER_BLOCK;     // 1024 blocks of 16 waves
    _KLD_MVNx2_kernel<<<blocks, THREADS_PER_BLOCK, 0, stream>>>(
        mu_q, sigma_q, mu_p, sigma_p, out);
}